// SyntaxAttentionGAT_87144886436421
// MI455X (gfx1250) — compile-verified
//
#include <hip/hip_runtime.h>
#include <hip/hip_bf16.h>

// ---------------------------------------------------------------------------
// Problem constants (from reference setup_inputs): B=4, S=128, D=768, O=256
// ---------------------------------------------------------------------------
#define BB 4
#define SS 128
#define DD 768
#define OO 256

typedef __attribute__((ext_vector_type(2))) float v2f;
typedef __attribute__((ext_vector_type(8))) float v8f;

// Exact parameter type of __builtin_amdgcn_global_load_async_to_lds_b128,
// per the compiler diagnostic: vector_size(16) int, AS1 (global) / AS3 (LDS).
typedef int v4i_vs __attribute__((vector_size(16)));
typedef __attribute__((address_space(1))) v4i_vs* as1_v4i_p;
typedef __attribute__((address_space(3))) v4i_vs* as3_v4i_p;

#if __has_builtin(__builtin_amdgcn_global_load_async_to_lds_b128)
#define HAVE_ASYNC_LDS 1
#else
#define HAVE_ASYNC_LDS 0
#endif

__device__ __forceinline__ float sigmoidf_(float x) {
    return 1.0f / (1.0f + __expf(-x));
}

// Async 16-byte global->LDS copy (falls back to synchronous copy when the
// builtin is unavailable; the wait below then becomes a no-op path).
__device__ __forceinline__ void async_copy16(const float* g, float* l) {
#if HAVE_ASYNC_LDS
    __builtin_amdgcn_global_load_async_to_lds_b128((as1_v4i_p)(g), (as3_v4i_p)(l), 0, 0);
#else
    *(float4*)l = *(const float4*)g;
#endif
}

__device__ __forceinline__ void wait_async0() {
#if HAVE_ASYNC_LDS
#if __has_builtin(__builtin_amdgcn_s_wait_asynccnt)
    __builtin_amdgcn_s_wait_asynccnt(0);
#else
    asm volatile("s_wait_asynccnt 0x0" ::: "memory");
#endif
#endif
}

// ---------------------------------------------------------------------------
// K0: dep_w = sigmoid(dep_matrix * wp + bp)         (B*S*S = 65536 elements)
// ---------------------------------------------------------------------------
__global__ void k_depw(const float* __restrict__ dep, const float* __restrict__ wp,
                       const float* __restrict__ bp, float* __restrict__ dep_w) {
    int i = blockIdx.x * blockDim.x + threadIdx.x;
    if (i < BB * SS * SS) dep_w[i] = sigmoidf_(dep[i] * wp[0] + bp[0]);
}

// ---------------------------------------------------------------------------
// K1: fused 5-point stencil (gether) + syntax scale.
// ---------------------------------------------------------------------------
__global__ void k_stencil(const float* __restrict__ h, const float* __restrict__ a_simi,
                          const float* __restrict__ t_simi, const float* __restrict__ dep_w,
                          float* __restrict__ h_sp, float* __restrict__ h_syn0) {
    const int D4 = DD / 4;
    long gid = (long)blockIdx.x * blockDim.x + threadIdx.x;
    int d4 = gid % D4;
    int j  = (gid / D4) % SS;
    int i  = (gid / D4 / SS) % SS;
    int b  = gid / D4 / SS / SS;

    const float4* hv = (const float4*)h;
    size_t base = (((size_t)b * SS + i) * SS + j) * D4 + d4;
    float4 c = hv[base];
    float4 acc = c;

    if (j > 0) {
        float t = t_simi[b * SS + (j - 1)];
        float4 v = hv[base - D4];
        acc.x += v.x * t; acc.y += v.y * t; acc.z += v.z * t; acc.w += v.w * t;
    }
    if (j < SS - 1) {
        float t = t_simi[b * SS + (j + 1)];
        float4 v = hv[base + D4];
        acc.x += v.x * t; acc.y += v.y * t; acc.z += v.z * t; acc.w += v.w * t;
    }
    if (i > 0) {
        float a = a_simi[b * SS + (i - 1)];
        float4 v = hv[base - (size_t)SS * D4];
        acc.x += v.x * a; acc.y += v.y * a; acc.z += v.z * a; acc.w += v.w * a;
    }
    if (i < SS - 1) {
        float a = a_simi[b * SS + (i + 1)];
        float4 v = hv[base + (size_t)SS * D4];
        acc.x += v.x * a; acc.y += v.y * a; acc.z += v.z * a; acc.w += v.w * a;
    }
    ((float4*)h_sp)[base] = acc;

    float w = dep_w[(b * SS + i) * SS + j];
    float4 s; s.x = c.x * w; s.y = c.y * w; s.z = c.z * w; s.w = c.w * w;
    ((float4*)h_syn0)[base] = s;
}

// ---------------------------------------------------------------------------
// K2: syntax einsum via WMMA f32 16x16x4:
//   h_sy[b,i,n] = h_syn0[b,i,n] + sum_k dep_w[b,i,k] * h_syn0[b,k,n]
// dep_w A-tile staged in LDS with +4 row padding (conflict-free ds_loads).
// ---------------------------------------------------------------------------
__global__ void k_syntax_gemm(const float* __restrict__ dep_w, const float* __restrict__ hs,
                              float* __restrict__ out) {
    const size_t NCOL = (size_t)SS * DD;           // 98304
    const int NTG = (SS * DD / 16) / 8;            // 768 tn-groups per (b,tm)
    constexpr int AST = SS + 4;                    // padded row stride (132)

    int blk = blockIdx.x;
    int tng = blk % NTG;
    int tm  = (blk / NTG) % (SS / 16);
    int b   = blk / NTG / (SS / 16);
    int m0  = tm * 16;

    __shared__ float As[16 * AST];                 // ~8.25 KB

    {   // cooperative load: each thread copies 8 contiguous floats of one row
        int t = threadIdx.x;
        int r = t >> 4, cc = (t & 15) * 8;
        const float4* s = (const float4*)(dep_w + ((size_t)b * SS + m0 + r) * SS + cc);
        *(float4*)&As[r * AST + cc]     = s[0];
        *(float4*)&As[r * AST + cc + 4] = s[1];
    }
    __syncthreads();

    int wave = threadIdx.x >> 5, lane = threadIdx.x & 31;
    int hi = lane >> 4, l16 = lane & 15;
    int n0 = (tng * 8 + wave) * 16;

    const float* Bp = hs + (size_t)b * SS * NCOL;  // [k][n], row stride NCOL

    v8f c;
    #pragma unroll
    for (int r = 0; r < 8; ++r)                    // C init = h_syn0 (the "+" term)
        c[r] = Bp[(size_t)(m0 + r + 8 * hi) * NCOL + n0 + l16];

    #pragma unroll 4
    for (int k0 = 0; k0 < SS; k0 += 4) {
        int ak = k0 + 2 * hi;
        v2f a, bf;
        a[0] = As[l16 * AST + ak];
        a[1] = As[l16 * AST + ak + 1];
        bf[0] = Bp[(size_t)ak * NCOL + n0 + l16];
        bf[1] = Bp[(size_t)(ak + 1) * NCOL + n0 + l16];
        c = __builtin_amdgcn_wmma_f32_16x16x4_f32(false, a, false, bf,
                                                  (short)0, c, false, false);
    }

    float* op = out + (size_t)b * SS * NCOL;
    #pragma unroll
    for (int r = 0; r < 8; ++r)
        op[(size_t)(m0 + r + 8 * hi) * NCOL + n0 + l16] = c[r];
}

// ---------------------------------------------------------------------------
// K3: combine.  One wave per (b,i,j) row (D=768 = 32 lanes x 24).
// ---------------------------------------------------------------------------
__global__ void k_combine(const float* h_sp, const float* __restrict__ h_sy,
                          const float* __restrict__ h_orig, const float* __restrict__ Wg,
                          const float* __restrict__ bg, float* out) {
    const int Q = DD / 32;                         // 24
    int row  = blockIdx.x * 8 + (threadIdx.x >> 5);
    int lane = threadIdx.x & 31;
    const float* sp = h_sp   + (size_t)row * DD;
    const float* sy = h_sy   + (size_t)row * DD;
    const float* ho = h_orig + (size_t)row * DD;

    float vsp[Q], vsy[Q];
    float dot = 0.f;
    #pragma unroll
    for (int q = 0; q < Q; ++q) {
        int d = lane + 32 * q;
        vsp[q] = sp[d];
        vsy[q] = sy[d];
        dot += vsp[q] * Wg[d] + vsy[q] * Wg[DD + d];
    }
    #pragma unroll
    for (int m = 16; m >= 1; m >>= 1) dot += __shfl_xor(dot, m, 32);

    float g = sigmoidf_(dot + bg[0]);
    float* o = out + (size_t)row * DD;
    #pragma unroll
    for (int q = 0; q < Q; ++q) {
        int d = lane + 32 * q;
        o[d] = g * vsp[q] + (1.f - g) * vsy[q] + ho[d];
    }
}

// ---------------------------------------------------------------------------
// K4: fused GEMM + bias + t5_norm(rms) + relu.
// X tiles double-buffered in LDS via GLOBAL_LOAD_ASYNC_TO_LDS_B128
// (ASYNCcnt + s_wait_asynccnt), padded row stride (conflict-free ds_loads).
// Row sums of squares via shfl_xor(width 16) + ds_add_f32.
// ---------------------------------------------------------------------------
template <int DIN, int DOUT>
__global__ void k_gemm_norm(const float* __restrict__ X, const float* __restrict__ W,
                            const float* __restrict__ bias, const float* __restrict__ gain,
                            float* __restrict__ Y) {
    constexpr int TPW = (DOUT / 16) / 8;           // N-tiles per wave: 6 or 2
    constexpr int KC  = 128;                       // K-chunk
    constexpr int XST = KC + 4;                    // padded LDS row stride

    __shared__ float Xs[2][16 * XST];              // 2 x 8.25 KB
    __shared__ float rowsum[16];

    int t = threadIdx.x;
    if (t < 16) rowsum[t] = 0.f;

    int wave = t >> 5, lane = t & 31, hi = lane >> 4, l16 = lane & 15;
    size_t row0 = (size_t)blockIdx.x * 16;

    // issue one 16 x KC tile copy (512 float4s, 2 per thread)
    auto issue_tile = [&](int kc, int bufi) {
        #pragma unroll
        for (int p = 0; p < 2; ++p) {
            int idx = t + p * 256;                 // float4 index within tile
            int r   = idx >> 5;                    // KC/4 = 32 float4s per row
            int c4  = (idx & 31) * 4;
            async_copy16(X + (row0 + r) * DIN + kc + c4, &Xs[bufi][r * XST + c4]);
        }
    };

    v8f c[TPW];
    #pragma unroll
    for (int q = 0; q < TPW; ++q) {
        v8f z = {0.f, 0.f, 0.f, 0.f, 0.f, 0.f, 0.f, 0.f};
        c[q] = z;
    }

    issue_tile(0, 0);
    int bufi = 0;
    for (int kc = 0; kc < DIN; kc += KC) {
        wait_async0();
        __syncthreads();                           // tile landed; prev compute done
        if (kc + KC < DIN) issue_tile(kc + KC, bufi ^ 1);

        const float* Xc = Xs[bufi];
        #pragma unroll 4
        for (int kk = 0; kk < KC; kk += 4) {
            int ak = kk + 2 * hi;
            v2f a;
            a[0] = Xc[l16 * XST + ak];
            a[1] = Xc[l16 * XST + ak + 1];
            int kg = kc + ak;
            #pragma unroll
            for (int q = 0; q < TPW; ++q) {
                int n0 = (wave * TPW + q) * 16;
                v2f bf;
                bf[0] = W[(size_t)kg * DOUT + n0 + l16];
                bf[1] = W[(size_t)(kg + 1) * DOUT + n0 + l16];
                c[q] = __builtin_amdgcn_wmma_f32_16x16x4_f32(false, a, false, bf,
                                                             (short)0, c[q], false, false);
            }
        }
        bufi ^= 1;
    }

    // bias + per-row sum of squares (partial per lane)
    float part[8];
    #pragma unroll
    for (int r = 0; r < 8; ++r) part[r] = 0.f;
    #pragma unroll
    for (int q = 0; q < TPW; ++q) {
        int n0 = (wave * TPW + q) * 16;
        float bv = bias[n0 + l16];
        #pragma unroll
        for (int r = 0; r < 8; ++r) { c[q][r] += bv; part[r] += c[q][r] * c[q][r]; }
    }
    #pragma unroll
    for (int r = 0; r < 8; ++r)
        #pragma unroll
        for (int m = 8; m >= 1; m >>= 1) part[r] += __shfl_xor(part[r], m, 16);
    if (l16 == 0) {
        #pragma unroll
        for (int r = 0; r < 8; ++r) atomicAdd(&rowsum[r + 8 * hi], part[r]);
    }
    __syncthreads();

    float inv[8];
    #pragma unroll
    for (int r = 0; r < 8; ++r)
        inv[r] = rsqrtf(rowsum[r + 8 * hi] / (float)DOUT + 1e-12f);

    #pragma unroll
    for (int q = 0; q < TPW; ++q) {
        int n0 = (wave * TPW + q) * 16;
        float gv = gain[n0 + l16];
        #pragma unroll
        for (int r = 0; r < 8; ++r) {
            float v = c[q][r] * inv[r] * gv;
            Y[(row0 + r + 8 * hi) * DOUT + n0 + l16] = v > 0.f ? v : 0.f;
        }
    }
}

// ---------------------------------------------------------------------------
// Launch: inputs in setup_inputs() order:
// 0 table, 1 a_simi, 2 t_simi, 3 dep_matrix, 4 W1, 5 b1, 6 W2, 7 b2,
// 8 wp, 9 bp, 10 Wg, 11 bg, 12 g1, 13 g2
// ---------------------------------------------------------------------------
extern "C" void kernel_launch(void* const* d_in, const int* in_sizes, int n_in,
                              void* d_out, int out_size, void* d_ws, size_t ws_size,
                              hipStream_t stream) {
    const float* table = (const float*)d_in[0];
    const float* a_s   = (const float*)d_in[1];
    const float* t_s   = (const float*)d_in[2];
    const float* depm  = (const float*)d_in[3];
    const float* W1    = (const float*)d_in[4];
    const float* b1    = (const float*)d_in[5];
    const float* W2    = (const float*)d_in[6];
    const float* b2    = (const float*)d_in[7];
    const float* wp    = (const float*)d_in[8];
    const float* bp    = (const float*)d_in[9];
    const float* Wg    = (const float*)d_in[10];
    const float* bg    = (const float*)d_in[11];
    const float* g1    = (const float*)d_in[12];
    const float* g2    = (const float*)d_in[13];
    float* out = (float*)d_out;

    const size_t TAB = (size_t)BB * SS * SS * DD;  // 50,331,648 floats
    float* ws    = (float*)d_ws;
    float* dep_w = ws;                             // 65536 floats
    float* buf1  = ws + 65536;                     // h_sp / combined (in place)
    float* buf2  = buf1 + TAB;                     // h_syn0
    float* buf3  = buf2 + TAB;                     // h_sy
    float* buf4  = buf3 + TAB;                     // h1

    const int thr = 256;
    dim3 blk(thr);

    k_depw<<<dim3((BB * SS * SS + thr - 1) / thr), blk, 0, stream>>>(depm, wp, bp, dep_w);

    const long n_sten = (long)BB * SS * SS * (DD / 4);
    const int g_sten  = (int)(n_sten / thr);                    // 49152
    const int g_syn   = BB * (SS / 16) * ((SS * DD / 16) / 8);  // 24576
    const int g_comb  = BB * SS * SS / 8;                       // 8192
    const int g_gemm  = BB * SS * SS / 16;                      // 4096

    // ---- Layer 1 ----
    k_stencil<<<dim3(g_sten), blk, 0, stream>>>(table, a_s, t_s, dep_w, buf1, buf2);
    k_syntax_gemm<<<dim3(g_syn), blk, 0, stream>>>(dep_w, buf2, buf3);
    k_combine<<<dim3(g_comb), blk, 0, stream>>>(buf1, buf3, table, Wg, bg, buf1);
    k_gemm_norm<DD, DD><<<dim3(g_gemm), blk, 0, stream>>>(buf1, W1, b1, g1, buf4);

    // ---- Layer 2 ----
    k_stencil<<<dim3(g_sten), blk, 0, stream>>>(buf4, a_s, t_s, dep_w, buf1, buf2);
    k_syntax_gemm<<<dim3(g_syn), blk, 0, stream>>>(dep_w, buf2, buf3);
    k_combine<<<dim3(g_comb), blk, 0, stream>>>(buf1, buf3, buf4, Wg, bg, buf1);
    k_gemm_norm<DD, OO><<<dim3(g_gemm), blk, 0, stream>>>(buf1, W2, b2, g2, out);
}